// SuperInteractionBlock_47974784696341
// MI455X (gfx1250) — compile-verified
//
#include <hip/hip_runtime.h>
#include <math.h>

typedef float v2f __attribute__((ext_vector_type(2)));
typedef float v8f __attribute__((ext_vector_type(8)));

#define HIDDEN 128
#define NFILT 64

// ---------------------------------------------------------------------------
// 1) Fold filter_net's second layer: fsum needs only rowsum(Wf2) and sum(bf2).
// ---------------------------------------------------------------------------
__global__ void prep_kernel(const float* __restrict__ Wf2, const float* __restrict__ bf2,
                            float* __restrict__ wsum, float* __restrict__ bsum) {
    int f = threadIdx.x;
    if (f < NFILT) {
        float s = 0.f;
        #pragma unroll 8
        for (int j = 0; j < NFILT; ++j) s += Wf2[f * NFILT + j];
        wsum[f] = s;
    }
    if (f == 0) {
        float s = 0.f;
        for (int j = 0; j < NFILT; ++j) s += bf2[j];
        *bsum = s;
    }
}

__global__ void zero_kernel(float4* __restrict__ p, int n4) {
    int i = blockIdx.x * blockDim.x + threadIdx.x;
    if (i < n4) p[i] = make_float4(0.f, 0.f, 0.f, 0.f);
}

// ---------------------------------------------------------------------------
// 2) Per-edge scalar filter: fsum[e] = cut(w) * (bsum + sum_f tanh(...)*wsum[f])
// ---------------------------------------------------------------------------
__global__ void edge_fsum_kernel(const float* __restrict__ ew, const float* __restrict__ Wf1,
                                 const float* __restrict__ bf1, const float* __restrict__ wsum,
                                 const float* __restrict__ bsum, float* __restrict__ fsum, int E) {
    int e = blockIdx.x * blockDim.x + threadIdx.x;
    if (e >= E) return;
    float w = ew[e];
    float scaled = fmaf(w, 0.25f, -1.0f);          // w * (2/CUTOFF) - 1
    float acc = *bsum;
    #pragma unroll 8
    for (int f = 0; f < NFILT; ++f)
        acc += tanhf(fmaf(scaled, Wf1[f], bf1[f])) * wsum[f];
    float cut = (w <= 8.0f) ? 0.5f * (__cosf(w * 0.39269908169872414f) + 1.0f) : 0.f;
    fsum[e] = acc * cut;
}

// ---------------------------------------------------------------------------
// 3) Gather x[col], scale by fsum, scatter-add into agg[row] (L2-resident f32
//    hardware atomics). 32 lanes per edge, float4 per lane.
// ---------------------------------------------------------------------------
__global__ void scatter_kernel(const float* __restrict__ x, const int* __restrict__ row,
                               const int* __restrict__ col, const float* __restrict__ fsum,
                               float* __restrict__ agg, int E) {
    int tid = blockIdx.x * blockDim.x + threadIdx.x;
    int e = tid >> 5;
    if (e >= E) return;
    int q = tid & 31;
    int r = row[e];
    int c = col[e];
    float f = fsum[e];
    float4 xv = ((const float4*)x)[c * (HIDDEN / 4) + q];
    float* dst = agg + (size_t)r * HIDDEN + q * 4;
    unsafeAtomicAdd(dst + 0, xv.x * f);
    unsafeAtomicAdd(dst + 1, xv.y * f);
    unsafeAtomicAdd(dst + 2, xv.z * f);
    unsafeAtomicAdd(dst + 3, xv.w * f);
}

// ---------------------------------------------------------------------------
// WMMA f32 16x16x4 strip: one wave computes D[16 x 128] for a 128-K GEMM.
// A layout (16x4 f32): lane = M (lane&15), half-wave selects K pair -> v2f.
// B layout (4x16 f32): lane = N, half-wave selects K pair -> v2f.
// D layout: n = lane&15, m = vgpr + 8*(lane>>4).
// ---------------------------------------------------------------------------
__device__ __forceinline__ void wmma_strip_f32(const float* __restrict__ A,
                                               const float* __restrict__ W,
                                               int rowBase, int N, int lane, v8f acc[8]) {
    int nm = lane & 15;
    int kh = (lane >> 4) << 1;                     // 0 or 2
    int arow = rowBase + nm;
    if (arow >= N) arow = N - 1;                   // clamp; stores are guarded
    const float* Arow = A + (size_t)arow * HIDDEN;
    for (int k = 0; k < HIDDEN; k += 4) {
        v2f a;
        a.x = Arow[k + kh];
        a.y = Arow[k + kh + 1];
        #pragma unroll
        for (int t = 0; t < 8; ++t) {
            int n = t * 16 + nm;
            v2f b;
            b.x = W[(size_t)(k + kh) * HIDDEN + n];
            b.y = W[(size_t)(k + kh + 1) * HIDDEN + n];
            acc[t] = __builtin_amdgcn_wmma_f32_16x16x4_f32(
                false, a, false, b, (short)0, acc[t], false, false);
        }
    }
}

// ---------------------------------------------------------------------------
// 4) hidden = softplus(agg @ Wi1 + bi1)
// ---------------------------------------------------------------------------
__global__ void gemm_softplus_kernel(const float* __restrict__ A, const float* __restrict__ W,
                                     const float* __restrict__ bias, float* __restrict__ out,
                                     int N) {
    int lane = threadIdx.x & 31;
    int wave = threadIdx.x >> 5;
    int rowBase = (blockIdx.x * (blockDim.x >> 5) + wave) * 16;
    if (rowBase >= N) return;
    v8f acc[8] = {};
    wmma_strip_f32(A, W, rowBase, N, lane, acc);
    int nm = lane & 15;
    int moff = (lane >> 4) * 8;
    #pragma unroll
    for (int t = 0; t < 8; ++t) {
        int n = t * 16 + nm;
        float bv = bias[n];
        #pragma unroll
        for (int v = 0; v < 8; ++v) {
            int r = rowBase + v + moff;
            if (r < N) {
                float val = acc[t][v] + bv;
                float sp = (val > 20.f) ? val : log1pf(__expf(val));
                out[(size_t)r * HIDDEN + n] = sp;
            }
        }
    }
}

// ---------------------------------------------------------------------------
// 5) out = BN(hidden @ Wi2 + bi2)   (inference-mode batchnorm)
// ---------------------------------------------------------------------------
__global__ void gemm_bn_kernel(const float* __restrict__ A, const float* __restrict__ W,
                               const float* __restrict__ bias, const float* __restrict__ gamma,
                               const float* __restrict__ beta, const float* __restrict__ mean,
                               const float* __restrict__ var, float* __restrict__ out, int N) {
    int lane = threadIdx.x & 31;
    int wave = threadIdx.x >> 5;
    int rowBase = (blockIdx.x * (blockDim.x >> 5) + wave) * 16;
    if (rowBase >= N) return;
    v8f acc[8] = {};
    wmma_strip_f32(A, W, rowBase, N, lane, acc);
    int nm = lane & 15;
    int moff = (lane >> 4) * 8;
    #pragma unroll
    for (int t = 0; t < 8; ++t) {
        int n = t * 16 + nm;
        float bv = bias[n];
        float scale = gamma[n] * rsqrtf(var[n] + 1e-3f);
        float shift = beta[n] - mean[n] * scale;
        #pragma unroll
        for (int v = 0; v < 8; ++v) {
            int r = rowBase + v + moff;
            if (r < N)
                out[(size_t)r * HIDDEN + n] = fmaf(acc[t][v] + bv, scale, shift);
        }
    }
}

// ---------------------------------------------------------------------------
extern "C" void kernel_launch(void* const* d_in, const int* in_sizes, int n_in,
                              void* d_out, int out_size, void* d_ws, size_t ws_size,
                              hipStream_t stream) {
    const float* x     = (const float*)d_in[0];
    const int*   eidx  = (const int*)  d_in[1];
    const float* ew    = (const float*)d_in[2];
    /* d_in[3] edge_attr is unused by the reference */
    const float* Wf1   = (const float*)d_in[4];
    const float* bf1   = (const float*)d_in[5];
    const float* Wf2   = (const float*)d_in[6];
    const float* bf2   = (const float*)d_in[7];
    const float* Wi1   = (const float*)d_in[8];
    const float* bi1   = (const float*)d_in[9];
    const float* Wi2   = (const float*)d_in[10];
    const float* bi2   = (const float*)d_in[11];
    const float* gamma = (const float*)d_in[12];
    const float* beta  = (const float*)d_in[13];
    const float* mean  = (const float*)d_in[14];
    const float* var   = (const float*)d_in[15];

    const int E = in_sizes[2];
    const int N = in_sizes[0] / HIDDEN;

    float* ws     = (float*)d_ws;
    float* wsum   = ws;                     // 64
    float* bsum   = ws + 64;                // 1
    float* fsum   = ws + 256;               // E
    float* agg    = fsum + E;               // N*128
    float* hidden = agg + (size_t)N * HIDDEN; // N*128
    float* out    = (float*)d_out;

    prep_kernel<<<1, 64, 0, stream>>>(Wf2, bf2, wsum, bsum);

    int n4 = N * HIDDEN / 4;
    zero_kernel<<<(n4 + 255) / 256, 256, 0, stream>>>((float4*)agg, n4);

    edge_fsum_kernel<<<(E + 255) / 256, 256, 0, stream>>>(ew, Wf1, bf1, wsum, bsum, fsum, E);

    long total = (long)E * 32;
    scatter_kernel<<<(int)((total + 255) / 256), 256, 0, stream>>>(
        x, eidx, eidx + E, fsum, agg, E);

    int rowTiles = (N + 15) / 16;
    int blocks = (rowTiles + 7) / 8;   // 8 waves (16-row strips) per 256-thread block
    gemm_softplus_kernel<<<blocks, 256, 0, stream>>>(agg, Wi1, bi1, hidden, N);
    gemm_bn_kernel<<<blocks, 256, 0, stream>>>(hidden, Wi2, bi2, gamma, beta, mean, var, out, N);
}